// PhyloNeighbours_59906203845290
// MI455X (gfx1250) — compile-verified
//
#include <hip/hip_runtime.h>

typedef __attribute__((ext_vector_type(2))) float v2f;
typedef __attribute__((ext_vector_type(8))) float v8f;

#define B_    1024
#define F_    4096
#define CD_   64
#define K_    16
#define OUTC_ (F_ * K_ + 1)

#define COLS_PER_BLOCK 512
#define LDS_STRIDE     520   // 512 + 8 floats pad -> row r starts at bank (8*r)%64

// ---------------------------------------------------------------------------
// Kernel 0: per-feature squared norms  sq[f] = sum_c coord[c][f]^2
// coord is (64, 4096) row-major.
// ---------------------------------------------------------------------------
__global__ void sq_kernel(const float* __restrict__ coord, float* __restrict__ sq) {
    int f = blockIdx.x * blockDim.x + threadIdx.x;
    float s = 0.0f;
#pragma unroll 8
    for (int c = 0; c < CD_; ++c) {
        float v = coord[c * F_ + f];
        s = fmaf(v, v, s);
    }
    sq[f] = s;
}

// Load one 4x16 B fragment set (all 16 K-steps) for column tile at j0.
// B layout guess mirrors A 16x4: VGPR0 {lanes0-15:K=k0, lanes16-31:K=k0+2},
// VGPR1 {K=k0+1 / k0+3}.
__device__ __forceinline__ void load_btile(v2f bv[16],
                                           const float* __restrict__ coord,
                                           int j0, int lo, int hi) {
#pragma unroll
    for (int kk = 0; kk < 16; ++kk) {
        int k0 = kk * 4 + 2 * hi;
        bv[kk].x = coord[(k0    ) * F_ + j0 + lo];
        bv[kk].y = coord[(k0 + 1) * F_ + j0 + lo];
    }
}

// ---------------------------------------------------------------------------
// Kernel 1: fused distance (WMMA f32 16x16x4) + per-row top-16 selection.
// One block = 16-row strip of the 4096x4096 distance matrix; 8 wave32.
// ---------------------------------------------------------------------------
__global__ void __launch_bounds__(256)
dist_topk_kernel(const float* __restrict__ coord,
                 const float* __restrict__ sq,
                 int* __restrict__ nbr) {
    // 16 * 520 * 4 = 33,280 bytes; candidate overlay needs 32,768 bytes.
    __shared__ __align__(16) unsigned char smem_raw[16 * LDS_STRIDE * 4];
    float* distS = (float*)smem_raw;
    unsigned long long* cand = (unsigned long long*)smem_raw;

    const int tid  = threadIdx.x;
    const int lane = tid & 31;
    const int wave = tid >> 5;
    const int lo   = lane & 15;
    const int hi   = lane >> 4;
    const int i0   = blockIdx.x * 16;

    // Preload A fragments for the whole strip: av[kk] covers K = 4*kk .. 4*kk+3.
    v2f av[16];
#pragma unroll
    for (int kk = 0; kk < 16; ++kk) {
        int k0 = kk * 4 + 2 * hi;
        av[kk].x = coord[(k0    ) * F_ + i0 + lo];
        av[kk].y = coord[(k0 + 1) * F_ + i0 + lo];
    }
    // sq for the 8 rows this lane's accumulator VGPRs touch (row = r + 8*hi).
    float sqi[8];
#pragma unroll
    for (int r = 0; r < 8; ++r) sqi[r] = sq[i0 + r + 8 * hi];

    // Register-resident sorted top-16 of keys (dist_bits<<32)|col, ascending.
    unsigned long long best[16];
#pragma unroll
    for (int t = 0; t < 16; ++t) best[t] = ~0ull;

    for (int cb = 0; cb < F_; cb += COLS_PER_BLOCK) {
        // ---- compute 512 columns of the distance strip into LDS ----
        // Tiles: tIdx = wave + q*8, q = 0..3. Double-buffer B across q so the
        // WMMA chain never waits on its own tile's loads back-to-back.
        v2f bcur[16];
        load_btile(bcur, coord, cb + wave * 16, lo, hi);
#pragma unroll
        for (int q = 0; q < 4; ++q) {
            int tIdx = wave + q * 8;           // 0..31
            int j0   = cb + tIdx * 16;
            v2f bnxt[16];
            if (q < 3) load_btile(bnxt, coord, j0 + 128, lo, hi);

            v8f acc = {};
#pragma unroll
            for (int kk = 0; kk < 16; ++kk) {
                acc = __builtin_amdgcn_wmma_f32_16x16x4_f32(
                        /*neg_a=*/false, av[kk], /*neg_b=*/false, bcur[kk],
                        /*c_mod=*/(short)0, acc, /*reuse_a=*/false, /*reuse_b=*/false);
            }

            float sqc      = sq[j0 + lo];
            int   colLocal = tIdx * 16 + lo;
#pragma unroll
            for (int r = 0; r < 8; ++r) {
                // reference order: (-2*G + sq[col]) + sq[row], clamp, sqrt
                float d = (-2.0f * acc[r] + sqc) + sqi[r];
                d = sqrtf(fmaxf(d, 0.0f));
                distS[(r + 8 * hi) * LDS_STRIDE + colLocal] = d;
            }
            if (q < 3) {
#pragma unroll
                for (int kk = 0; kk < 16; ++kk) bcur[kk] = bnxt[kk];
            }
        }
        __syncthreads();

        // ---- scan: thread owns (row = tid>>4, 32-column segment = tid&15) ----
        {
            int row  = tid >> 4;
            int seg  = tid & 15;
            int base = seg * 32;
            for (int c = 0; c < 32; ++c) {
                int   colLocal = base + c;
                float d        = distS[row * LDS_STRIDE + colLocal];
                unsigned long long key =
                    ((unsigned long long)__float_as_uint(d) << 32) |
                    (unsigned)(cb + colLocal);
                if (key < best[15]) {
                    best[15] = key;
#pragma unroll
                    for (int t = 15; t > 0; --t) {
                        if (best[t] < best[t - 1]) {
                            unsigned long long tmp = best[t];
                            best[t] = best[t - 1];
                            best[t - 1] = tmp;
                        }
                    }
                }
            }
        }
        __syncthreads();   // LDS reused next iteration
    }

    // ---- merge: dump per-thread lists (overlay on dist LDS), one thread/row ----
#pragma unroll
    for (int t = 0; t < 16; ++t) cand[tid * 16 + t] = best[t];
    __syncthreads();

    if (tid < 16) {
        int row = tid;
        unsigned long long fin[16];
#pragma unroll
        for (int t = 0; t < 16; ++t) fin[t] = ~0ull;
        for (int m = 0; m < 256; ++m) {
            unsigned long long key = cand[row * 256 + m];
            if (key < fin[15]) {
                fin[15] = key;
#pragma unroll
                for (int t = 15; t > 0; --t) {
                    if (fin[t] < fin[t - 1]) {
                        unsigned long long tmp = fin[t];
                        fin[t] = fin[t - 1];
                        fin[t - 1] = tmp;
                    }
                }
            }
        }
#pragma unroll
        for (int t = 0; t < 16; ++t)
            nbr[(i0 + row) * K_ + t] = (int)(fin[t] & 0xFFFFFFFFu);
    }
}

// ---------------------------------------------------------------------------
// Kernel 2: bandwidth phase. out[b][c] = inputs[b][c==0 ? 0 : nbr[c-1]]
// ---------------------------------------------------------------------------
__global__ void gather_kernel(const float* __restrict__ inputs,
                              const int* __restrict__ nbr,
                              float* __restrict__ out) {
    int c = blockIdx.x * blockDim.x + threadIdx.x;
    int b = blockIdx.y;
    if (c >= OUTC_) return;
    int src = (c == 0) ? 0 : nbr[c - 1];
    out[(long)b * OUTC_ + c] = inputs[b * F_ + src];
}

extern "C" void kernel_launch(void* const* d_in, const int* in_sizes, int n_in,
                              void* d_out, int out_size, void* d_ws, size_t ws_size,
                              hipStream_t stream) {
    const float* inputs = (const float*)d_in[0];  // (1024, 4096)
    const float* coord  = (const float*)d_in[1];  // (64, 4096)
    float* out = (float*)d_out;                   // (1024, 65537)

    float* sq  = (float*)d_ws;                               // 4096 f32
    int*   nbr = (int*)((char*)d_ws + F_ * sizeof(float));   // 4096*16 i32

    sq_kernel<<<F_ / 256, 256, 0, stream>>>(coord, sq);
    dist_topk_kernel<<<F_ / 16, 256, 0, stream>>>(coord, sq, nbr);

    dim3 g((OUTC_ + 255) / 256, B_);
    gather_kernel<<<g, 256, 0, stream>>>(inputs, nbr, out);
}